// SlowSSM_51745765982536
// MI455X (gfx1250) — compile-verified
//
#include <hip/hip_runtime.h>
#include <hip/hip_bf16.h>
#include <math.h>

// ---------------------------------------------------------------------------
// Mamba-style SSM block ×2 layers, MI455X (gfx1250, wave32, WMMA).
// Dims: B*N = 64 sequences, T = 256, D_MODEL = 128, D_INNER = 256,
//       D_STATE = 16, DT_RANK = 8, D_CONV = 4.
// All math in fp32 using V_WMMA_F32_16X16X4_F32 (exact-precision WMMA path).
// ---------------------------------------------------------------------------

typedef __attribute__((ext_vector_type(2))) float v2f;
typedef __attribute__((ext_vector_type(8))) float v8f;

#define SEQS     64
#define TLEN     256
#define DMODEL   128
#define DINNER   256
#define DSTATE   16
#define ROWS     (SEQS * TLEN)          // 16384
#define DBL_LD   48                     // padded width of [dt|B|C] buffer
#define XP_NPAD  48                     // x_proj_w padded from 40 -> 48 cols

#define EPI_NONE          0
#define EPI_SOFTPLUS_BIAS 1

// Store one 16x16 D tile (this lane's column, 8 rows) with optional epilogue.
__device__ __forceinline__ void store_tile(const v8f& acc, float* __restrict__ C,
                                           int mrow0, int ldc, int ncol,
                                           float b, int epi)
{
#pragma unroll
    for (int r = 0; r < 8; ++r) {
        float v = acc[r];
        if (epi == EPI_SOFTPLUS_BIAS) {
            v += b;
            v = (v > 20.0f) ? v : log1pf(__expf(v));   // softplus
        }
        C[(size_t)(mrow0 + r) * ldc + ncol] = v;
    }
}

// ---------------------------------------------------------------------------
// FP32 WMMA GEMM, register-blocked 4 m-tiles per wave (64x16 output / wave):
//   C[M x (16*tiles_n)] = A[M x K] @ W[K x 16*tiles_n]
// K multiple of 4, M multiple of 64, W fully padded (no column guard) so the
// whole wave runs uniform control flow -> EXEC all-1s at every WMMA.
// Per K-step: 2 scalar B loads amortized over 4 WMMAs, A as float2 loads.
// ---------------------------------------------------------------------------
__global__ __launch_bounds__(256) void wmma_gemm_f32_kernel(
    const float* __restrict__ A, const float* __restrict__ W,
    float* __restrict__ C, const float* __restrict__ bias,
    int M, int K, int tiles_n,
    int lda, int ldb, int ldc, int epi)
{
    const int wave = blockIdx.x * (blockDim.x >> 5) + (threadIdx.x >> 5);
    const int total_waves = (M >> 6) * tiles_n;         // 4 m-tiles per wave
    if (wave >= total_waves) return;                    // uniform per wave

    const int tm4  = wave / tiles_n;                    // 64-row block index
    const int tn   = wave % tiles_n;
    const int lane = threadIdx.x & 31;
    const int half = lane >> 4;                         // 0: lanes 0-15, 1: lanes 16-31
    const int l16  = lane & 15;

    const int ncol = (tn << 4) + l16;                   // B/D column for this lane
    const float* Abase = A + (size_t)((tm4 << 6) + l16) * lda;

    v8f acc0 = {}, acc1 = {}, acc2 = {}, acc3 = {};
    for (int k0 = 0; k0 < K; k0 += 4) {
        const int ka = k0 + (half << 1);                // halves hold k,k+1 / k+2,k+3
        v2f bf;
        bf.x = W[(size_t)ka * ldb + ncol];
        bf.y = W[(size_t)(ka + 1) * ldb + ncol];

        v2f a0, a1, a2, a3;
        a0.x = Abase[ka];                    a0.y = Abase[ka + 1];
        a1.x = Abase[(size_t)16 * lda + ka]; a1.y = Abase[(size_t)16 * lda + ka + 1];
        a2.x = Abase[(size_t)32 * lda + ka]; a2.y = Abase[(size_t)32 * lda + ka + 1];
        a3.x = Abase[(size_t)48 * lda + ka]; a3.y = Abase[(size_t)48 * lda + ka + 1];

        acc0 = __builtin_amdgcn_wmma_f32_16x16x4_f32(false, a0, false, bf, (short)0, acc0, false, false);
        acc1 = __builtin_amdgcn_wmma_f32_16x16x4_f32(false, a1, false, bf, (short)0, acc1, false, false);
        acc2 = __builtin_amdgcn_wmma_f32_16x16x4_f32(false, a2, false, bf, (short)0, acc2, false, false);
        acc3 = __builtin_amdgcn_wmma_f32_16x16x4_f32(false, a3, false, bf, (short)0, acc3, false, false);
    }

    const float b = (epi == EPI_SOFTPLUS_BIAS) ? bias[ncol] : 0.0f;
    const int mbase = (tm4 << 6) + (half << 3);         // D layout: reg r -> M=r (+8 hi half)
    store_tile(acc0, C, mbase +  0, ldc, ncol, b, epi);
    store_tile(acc1, C, mbase + 16, ldc, ncol, b, epi);
    store_tile(acc2, C, mbase + 32, ldc, ncol, b, epi);
    store_tile(acc3, C, mbase + 48, ldc, ncol, b, epi);
}

// ---------------------------------------------------------------------------
// Pad x_proj_w [256 x 40] -> [256 x 48] (zero-filled) so the GEMM needs no
// per-lane column guard (keeps EXEC uniform around WMMA).
// ---------------------------------------------------------------------------
__global__ __launch_bounds__(256) void pad_w_kernel(
    const float* __restrict__ w, float* __restrict__ wp, int total)
{
    const int i = blockIdx.x * blockDim.x + threadIdx.x;
    if (i >= total) return;
    const int k = i / XP_NPAD, n = i - k * XP_NPAD;
    wp[i] = (n < 40) ? w[k * 40 + n] : 0.0f;
}

// ---------------------------------------------------------------------------
// Causal depthwise conv (k=4) + bias + SiLU over the u-half of xz.
// xz: [ROWS x 512] (cols 0..255 = u_pre, 256..511 = res). u out: [ROWS x 256].
// ---------------------------------------------------------------------------
__global__ __launch_bounds__(256) void conv_silu_kernel(
    const float* __restrict__ xz, const float* __restrict__ cw,
    const float* __restrict__ cb, float* __restrict__ u, int total)
{
    const int i = blockIdx.x * blockDim.x + threadIdx.x;
    if (i >= total) return;
    const int d   = i & (DINNER - 1);
    const int row = i >> 8;                 // seq*T + t
    const int t   = row & (TLEN - 1);

    const float w0 = cw[d * 4 + 0], w1 = cw[d * 4 + 1];
    const float w2 = cw[d * 4 + 2], w3 = cw[d * 4 + 3];
    const float* base = xz + (size_t)row * 512 + d;

    float acc = cb[d] + w3 * base[0];
    if (t >= 1) acc += w2 * base[-1 * 512];
    if (t >= 2) acc += w1 * base[-2 * 512];
    if (t >= 3) acc += w0 * base[-3 * 512];

    u[(size_t)i] = acc / (1.0f + __expf(-acc));        // SiLU
}

// ---------------------------------------------------------------------------
// Selective scan + gate. One block per sequence, one thread per channel d.
// State h[16] lives in registers. The per-timestep B/C row (32 floats shared
// by all 256 threads) is staged through LDS. Fuses y = scan + u*D and the
// y*silu(res) gate; writes gated output in-place over u.
// ---------------------------------------------------------------------------
__global__ __launch_bounds__(256) void scan_gate_kernel(
    float* __restrict__ u,                 // [ROWS x 256] in, gated y out (in-place)
    const float* __restrict__ delta,       // [ROWS x 256]
    const float* __restrict__ dbl,         // [ROWS x 48]: cols 8..23 = B, 24..39 = C
    const float* __restrict__ xz,          // [ROWS x 512]: cols 256.. = res
    const float* __restrict__ A_log,       // [256 x 16]
    const float* __restrict__ Dp)          // [256]
{
    __shared__ float sBC[32];
    const int seq = blockIdx.x;
    const int d   = threadIdx.x;

    float a[DSTATE];
#pragma unroll
    for (int n = 0; n < DSTATE; ++n) a[n] = -__expf(A_log[d * DSTATE + n]);
    const float Dpar = Dp[d];

    float h[DSTATE];
#pragma unroll
    for (int n = 0; n < DSTATE; ++n) h[n] = 0.0f;

    for (int t = 0; t < TLEN; ++t) {
        const size_t row = (size_t)seq * TLEN + t;

        if (threadIdx.x < 32)              // stage B|C (dbl cols 8..39) into LDS
            sBC[threadIdx.x] = dbl[row * DBL_LD + 8 + threadIdx.x];
        if (t + 1 < TLEN)
            __builtin_prefetch(dbl + (row + 1) * DBL_LD, 0, 3);  // global_prefetch_b8

        const float dv = delta[row * DINNER + d];
        const float uv = u[row * DINNER + d];
        __syncthreads();

        const float du = dv * uv;
        float y = 0.0f;
#pragma unroll
        for (int n = 0; n < DSTATE; ++n) {
            const float dA = __expf(dv * a[n]);
            h[n] = dA * h[n] + du * sBC[n];             // ds_load broadcast
            y += h[n] * sBC[16 + n];
        }
        y += uv * Dpar;

        const float rv = xz[row * 512 + DINNER + d];
        const float gate = rv / (1.0f + __expf(-rv));   // silu(res)
        u[row * DINNER + d] = y * gate;
        __syncthreads();                                // protect sBC before next stage
    }
}

// ---------------------------------------------------------------------------
// Host-side orchestration
// ---------------------------------------------------------------------------
extern "C" void kernel_launch(void* const* d_in, const int* in_sizes, int n_in,
                              void* d_out, int out_size, void* d_ws, size_t ws_size,
                              hipStream_t stream)
{
    const float* x      = (const float*)d_in[0];  // [64,256,128]
    const float* in_w   = (const float*)d_in[1];  // [2,128,512]
    const float* conv_w = (const float*)d_in[2];  // [2,256,1,4]
    const float* conv_b = (const float*)d_in[3];  // [2,256]
    const float* xp_w   = (const float*)d_in[4];  // [2,256,40]
    const float* dt_w   = (const float*)d_in[5];  // [2,8,256]
    const float* dt_b   = (const float*)d_in[6];  // [2,256]
    const float* A_log  = (const float*)d_in[7];  // [2,256,16]
    const float* D_par  = (const float*)d_in[8];  // [2,256]
    const float* out_w  = (const float*)d_in[9];  // [2,256,128]
    float* out = (float*)d_out;                   // [64,256,128]

    float* ws = (float*)d_ws;
    size_t off = 0;
    float* xz    = ws + off;  off += (size_t)ROWS * 512;     // in_proj output
    float* ubuf  = ws + off;  off += (size_t)ROWS * DINNER;  // conv/silu u, then gated y
    float* dbl   = ws + off;  off += (size_t)ROWS * DBL_LD;  // [dt|B|C] padded to 48
    float* delta = ws + off;  off += (size_t)ROWS * DINNER;
    float* hbuf  = ws + off;  off += (size_t)ROWS * DMODEL;  // inter-layer activations
    float* wpad  = ws + off;  off += (size_t)DINNER * XP_NPAD; // padded x_proj_w

    const int M = ROWS;                  // 16384 rows, 256 64-row blocks
    const int mb = M >> 6;

    for (int l = 0; l < 2; ++l) {
        const float* hin  = (l == 0) ? x : hbuf;
        float* hout = (l == 0) ? hbuf : out;

        // 0) zero-pad x_proj_w: [256x40] -> [256x48]
        {
            int total = DINNER * XP_NPAD;
            pad_w_kernel<<<(total + 255) / 256, 256, 0, stream>>>(
                xp_w + (size_t)l * DINNER * 40, wpad, total);
        }
        // 1) xz = h @ in_proj_w   [16384x128]@[128x512]
        {
            int tiles_n = 512 / 16, waves = mb * tiles_n;
            wmma_gemm_f32_kernel<<<(waves + 7) / 8, 256, 0, stream>>>(
                hin, in_w + (size_t)l * DMODEL * 512, xz, nullptr,
                M, DMODEL, tiles_n, DMODEL, 512, 512, EPI_NONE);
        }
        // 2) u = silu(causal_dwconv(xz[:, :256]) + cb)
        {
            int total = ROWS * DINNER;
            conv_silu_kernel<<<total / 256, 256, 0, stream>>>(
                xz, conv_w + (size_t)l * DINNER * 4, conv_b + (size_t)l * DINNER,
                ubuf, total);
        }
        // 3) dbl = u @ x_proj_w_pad   [16384x256]@[256x48]
        {
            int tiles_n = XP_NPAD / 16, waves = mb * tiles_n;
            wmma_gemm_f32_kernel<<<(waves + 7) / 8, 256, 0, stream>>>(
                ubuf, wpad, dbl, nullptr,
                M, DINNER, tiles_n, DINNER, XP_NPAD, DBL_LD, EPI_NONE);
        }
        // 4) delta = softplus(dt @ dt_proj_w + dt_b)   [16384x8]@[8x256]
        {
            int tiles_n = DINNER / 16, waves = mb * tiles_n;
            wmma_gemm_f32_kernel<<<(waves + 7) / 8, 256, 0, stream>>>(
                dbl, dt_w + (size_t)l * 8 * DINNER, delta, dt_b + (size_t)l * DINNER,
                M, 8, tiles_n, DBL_LD, DINNER, DINNER, EPI_SOFTPLUS_BIAS);
        }
        // 5) selective scan + D skip + silu(res) gate (in-place into ubuf)
        scan_gate_kernel<<<SEQS, DINNER, 0, stream>>>(
            ubuf, delta, dbl, xz,
            A_log + (size_t)l * DINNER * DSTATE, D_par + (size_t)l * DINNER);

        // 6) h_out = y_gated @ out_proj_w   [16384x256]@[256x128]
        {
            int tiles_n = DMODEL / 16, waves = mb * tiles_n;
            wmma_gemm_f32_kernel<<<(waves + 7) / 8, 256, 0, stream>>>(
                ubuf, out_w + (size_t)l * DINNER * DMODEL, hout, nullptr,
                M, DINNER, tiles_n, DINNER, DMODEL, DMODEL, EPI_NONE);
        }
    }
    (void)in_sizes; (void)n_in; (void)out_size; (void)ws_size;
}